// DualPathSoftMoE2_3719441678627
// MI455X (gfx1250) — compile-verified
//
#include <hip/hip_runtime.h>
#include <hip/hip_bf16.h>
#include <math.h>

typedef __attribute__((ext_vector_type(2))) float v2f;
typedef __attribute__((ext_vector_type(4))) float v4f;
typedef __attribute__((ext_vector_type(8))) float v8f;

// ---------------------------------------------------------------------------
// D = A(16x4 f32) * B(4x16 f32) + C(16x16 f32), wave32 WMMA.
// Fragment maps (wave32):
//   A: lane L -> (M = L&15, K = 2*(L>>4) + v), v in {0,1}  (v2f)
//   B: lane L -> (N = L&15, K = 2*(L>>4) + v)              (v2f)
//   C/D: elem i, lane L -> (M = i + 8*(L>>4), N = L&15)    (v8f)
// ---------------------------------------------------------------------------
__device__ inline v8f wmma4(v2f a, v2f b, v8f c) {
  return __builtin_amdgcn_wmma_f32_16x16x4_f32(false, a, false, b, (short)0, c,
                                               false, false);
}

// ---------------------------------------------------------------------------
// Kernel 1: q = l2norm(layernorm(phi*qg+qb)) for the 16 core slots, stored
// pre-swizzled into the WMMA B-fragment layout:
//   qB[(d>>2)*64 + (j + 16*((d>>1)&1))*2 + (d&1)] = q[j][d]
// so the routing GEMM reads one coalesced b64 per lane per K-step.
// ---------------------------------------------------------------------------
__device__ inline float block_sum(float v, float* red, int tid) {
#pragma unroll
  for (int o = 16; o > 0; o >>= 1) v += __shfl_xor(v, o, 32);
  __syncthreads();
  if ((tid & 31) == 0) red[tid >> 5] = v;
  __syncthreads();
  float s = red[0];
#pragma unroll
  for (int k = 1; k < 8; ++k) s += red[k];
  return s;
}

__global__ __launch_bounds__(256) void qnorm_kernel(
    const float* __restrict__ phi, const float* __restrict__ qg,
    const float* __restrict__ qb, const float* __restrict__ lng,
    const float* __restrict__ lnb, float* __restrict__ qB) {
  __shared__ float red[8];
  const int j = blockIdx.x;  // core slot 0..15 == phi row j (phi is [E,S,D])
  const int tid = threadIdx.x;
  const int d0 = tid * 4;

  v4f ph = *(const v4f*)(phi + (size_t)j * 1024 + d0);
  v4f g = *(const v4f*)(qg + d0);
  v4f bb = *(const v4f*)(qb + d0);
  v4f t = ph * g + bb;

  float sum = block_sum(t.x + t.y + t.z + t.w, red, tid);
  float sums =
      block_sum(t.x * t.x + t.y * t.y + t.z * t.z + t.w * t.w, red, tid);
  float mean = sum * (1.0f / 1024.0f);
  float var = sums * (1.0f / 1024.0f) - mean * mean;
  float rstd = rsqrtf(var + 1e-5f);

  v4f lg = *(const v4f*)(lng + d0);
  v4f lb = *(const v4f*)(lnb + d0);
  v4f u = (t - mean) * rstd * lg + lb;

  float ss = block_sum(u.x * u.x + u.y * u.y + u.z * u.z + u.w * u.w, red, tid);
  float rn = 1.0f / (sqrtf(ss) + 1e-6f);
  u = u * rn;

#pragma unroll
  for (int q4 = 0; q4 < 4; ++q4) {
    int d = d0 + q4;
    int idx = (d >> 2) * 64 + (j + 16 * ((d >> 1) & 1)) * 2 + (d & 1);
    qB[idx] = u[q4];
  }
}

// ---------------------------------------------------------------------------
// Kernel 2: routing. 64 tokens/block, 4 waves; each wave owns 16 tokens.
// Per K-chunk of 128 columns (32 WMMA steps): stage y = x*kg+kb into LDS in
// A-fragment layout (stride 66 floats -> conflict-free writes & b64 reads),
// accumulate per-row sum(y^2), run 32 v_wmma_f32_16x16x4_f32.
// Norm is factored out of the GEMM (logit = acc * 1/(||y||+eps)).
// Then per token: softmax over 8 experts (lanes of equal parity) and exact
// entmax15 over 16 slots via bitonic sort + scans + ballot.
// ---------------------------------------------------------------------------
#define K2_WAVES 4
#define K2_CH 32
#define K2_FSTR 66

__global__ __launch_bounds__(128) void route_kernel(
    const float* __restrict__ x, const float* __restrict__ qB,
    const float* __restrict__ kg, const float* __restrict__ kb,
    const float* __restrict__ s0p, const float* __restrict__ s1p,
    float* __restrict__ dispatch, float* __restrict__ combine) {
  __shared__ float lds[K2_WAVES * K2_CH * K2_FSTR];
  const int tid = threadIdx.x;
  const int w = tid >> 5, lane = tid & 31;
  const int l16 = lane & 15, hi = lane >> 4;
  float* frW = lds + w * (K2_CH * K2_FSTR);
  const long long tok0 = (long long)blockIdx.x * 64 + w * 16;

  float ssq[16];
#pragma unroll
  for (int r = 0; r < 16; ++r) ssq[r] = 0.0f;

  v8f acc = {};
  for (int c = 0; c < 8; ++c) {
    const int col4 = (c * 32 + lane) * 4;
    v4f g = *(const v4f*)(kg + col4);
    v4f bb = *(const v4f*)(kb + col4);
#pragma unroll
    for (int r = 0; r < 16; ++r) {
      const float* xr = x + (size_t)(tok0 + r) * 1024;
      v4f xv = *(const v4f*)(xr + col4);
      v4f y = xv * g + bb;
      ssq[r] += y.x * y.x + y.y * y.y + y.z * y.z + y.w * y.w;
      float* p = frW + lane * K2_FSTR + r * 2;
      v2f lo; lo.x = y.x; lo.y = y.y;
      v2f hi2; hi2.x = y.z; hi2.y = y.w;
      *(v2f*)p = lo;          // fragLane = r      (K = 4k+{0,1})
      *(v2f*)(p + 32) = hi2;  // fragLane = r + 16 (K = 4k+{2,3})
    }
    for (int kl = 0; kl < K2_CH; ++kl) {
      v2f a = *(const v2f*)(frW + kl * K2_FSTR + lane * 2);
      v2f b = *(const v2f*)(qB + ((c * 32 + kl) * 32 + lane) * 2);
      acc = wmma4(a, b, acc);
    }
  }

#pragma unroll
  for (int r = 0; r < 16; ++r) {
#pragma unroll
    for (int o = 16; o > 0; o >>= 1) ssq[r] += __shfl_xor(ssq[r], o, 32);
  }

  const float inv0 = 1.0f / s0p[0];
  const float inv1h = 0.5f / s1p[0];

#pragma unroll
  for (int i = 0; i < 8; ++i) {
    float tot = hi ? ssq[i + 8] : ssq[i];
    float rn = 1.0f / (sqrtf(tot) + 1e-6f);
    float lgt = acc[i] * rn;  // logit for (token = i+8*hi, slot j = l16)

    // dispatch: softmax over 8 experts sharing slot parity (bit0 of j)
    float z0 = lgt * inv0;
    float m = z0;
    m = fmaxf(m, __shfl_xor(m, 2, 32));
    m = fmaxf(m, __shfl_xor(m, 4, 32));
    m = fmaxf(m, __shfl_xor(m, 8, 32));
    float ex = expf(z0 - m);
    float sm = ex;
    sm += __shfl_xor(sm, 2, 32);
    sm += __shfl_xor(sm, 4, 32);
    sm += __shfl_xor(sm, 8, 32);
    float disp = ex / sm;

    // combine: exact 1.5-entmax over the 16 core slots
    float z = lgt * inv1h;  // (l/scale1)/2
    float zm = z;
    zm = fmaxf(zm, __shfl_xor(zm, 1, 32));
    zm = fmaxf(zm, __shfl_xor(zm, 2, 32));
    zm = fmaxf(zm, __shfl_xor(zm, 4, 32));
    zm = fmaxf(zm, __shfl_xor(zm, 8, 32));
    z -= zm;

    float v = z;  // bitonic sort, descending, within each 16-lane half
#pragma unroll
    for (int k = 2; k <= 16; k <<= 1) {
#pragma unroll
      for (int j = k >> 1; j > 0; j >>= 1) {
        float o = __shfl_xor(v, j, 32);
        bool up = ((l16 & k) != 0);
        bool lower = ((l16 & j) == 0);
        v = (up == lower) ? fminf(v, o) : fmaxf(v, o);
      }
    }
    float cs = v, cq = v * v;  // inclusive scans over sorted values
#pragma unroll
    for (int dd = 1; dd < 16; dd <<= 1) {
      float t1 = __shfl_up(cs, dd, 16);
      float t2 = __shfl_up(cq, dd, 16);
      if (l16 >= dd) { cs += t1; cq += t2; }
    }
    float kf = (float)(l16 + 1);
    float mean = cs / kf;
    float msq = cq / kf;
    float ssv = kf * (msq - mean * mean);
    float delta = fmaxf((1.0f - ssv) / kf, 1e-12f);
    float tau = mean - sqrtf(delta);
    unsigned long long bal = __ballot(tau <= v);
    int support = __popc((unsigned)((bal >> (hi * 16)) & 0xFFFFu));
    float tauStar = __shfl(tau, hi * 16 + support - 1, 32);
    float pz = fmaxf(z - tauStar, 0.0f);
    float comb = pz * pz;

    long long token = tok0 + i + 8 * hi;
    dispatch[token * 16 + l16] = disp;
    combine[token * 16 + l16] = comb;
  }
}

// ---------------------------------------------------------------------------
// Kernel 3: slots partials: part[sp][b][j][d] = sum_{t in split} disp[t][j]*x[t][d]
// split-K=4 over tokens for occupancy; A = dispatch^T, B = x (both L2-hot).
// ---------------------------------------------------------------------------
__global__ __launch_bounds__(256) void slots_kernel(
    const float* __restrict__ x, const float* __restrict__ dispatch,
    float* __restrict__ part) {
  const int b = blockIdx.x, dc = blockIdx.y, sp = blockIdx.z;
  const int w = threadIdx.x >> 5, lane = threadIdx.x & 31;
  const int l16 = lane & 15, hi = lane >> 4;
  const int dBase = dc * 128 + w * 16;
  const float* xb = x + (size_t)b * 4096 * 1024;
  const float* db = dispatch + (size_t)b * 4096 * 16;
  v8f acc = {};
  const int t0base = sp * 1024;
  for (int kk = 0; kk < 256; ++kk) {
    int t0 = t0base + kk * 4 + 2 * hi;
    v2f a, bv;
    a.x = db[(size_t)t0 * 16 + l16];
    a.y = db[(size_t)(t0 + 1) * 16 + l16];
    bv.x = xb[(size_t)t0 * 1024 + dBase + l16];
    bv.y = xb[(size_t)(t0 + 1) * 1024 + dBase + l16];
    acc = wmma4(a, bv, acc);
  }
  float* out = part + (size_t)sp * (8 * 16 * 1024) + (size_t)b * 16 * 1024;
#pragma unroll
  for (int i = 0; i < 8; ++i)
    out[(size_t)(i + 8 * hi) * 1024 + dBase + l16] = acc[i];
}

__global__ void sum4_kernel(const float* __restrict__ p, float* __restrict__ o,
                            int n) {
  int i = blockIdx.x * blockDim.x + threadIdx.x;
  if (i < n) o[i] = p[i] + p[n + i] + p[2 * n + i] + p[3 * n + i];
}

// ---------------------------------------------------------------------------
// Kernel 5: H[e][r][h] = gelu(slots_e[r][:] . w1[e][:,h] + b1[e][h]).
// Rows r = 2*b + s -> slots[(r>>1)*16 + 2e + (r&1)].  Streams 128MB of w1
// (memory bound); prefetch (global_prefetch_b8) runs ahead of the K loop.
// ---------------------------------------------------------------------------
__global__ __launch_bounds__(256) void mlp1_kernel(
    const float* __restrict__ slots, const float* __restrict__ w1,
    const float* __restrict__ b1, float* __restrict__ H) {
  const int e = blockIdx.x, hc = blockIdx.y;
  const int w = threadIdx.x >> 5, lane = threadIdx.x & 31;
  const int l16 = lane & 15, hi = lane >> 4;
  const int h = hc * 128 + w * 16 + l16;
  const float* arow =
      slots + (size_t)((l16 >> 1) * 16 + e * 2 + (l16 & 1)) * 1024;
  const float* wb = w1 + (size_t)e * 1024 * 4096;
  v8f acc = {};
  for (int kk = 0; kk < 256; ++kk) {
    int k0 = kk * 4 + 2 * hi;
    v2f a = *(const v2f*)(arow + k0);
    v2f bv;
    bv.x = wb[(size_t)k0 * 4096 + h];
    bv.y = wb[(size_t)(k0 + 1) * 4096 + h];
    if (kk < 252) __builtin_prefetch(wb + (size_t)(k0 + 16) * 4096 + h, 0, 1);
    acc = wmma4(a, bv, acc);
  }
  float bias = b1[e * 4096 + h];
  float* Hout = H + (size_t)e * 16 * 4096;
#pragma unroll
  for (int i = 0; i < 8; ++i) {
    float vv = acc[i] + bias;
    float gl = 0.5f * vv * (1.0f + erff(vv * 0.70710678118654752f));
    Hout[(size_t)(i + 8 * hi) * 4096 + h] = gl;
  }
}

// ---------------------------------------------------------------------------
// Kernel 6: Ypart[sp][e][r][d] = H[e][r][:] . w2[e][:,d] over K-split of 1024.
// ---------------------------------------------------------------------------
__global__ __launch_bounds__(256) void mlp2_kernel(const float* __restrict__ H,
                                                   const float* __restrict__ w2,
                                                   float* __restrict__ part) {
  const int e = blockIdx.x, dc = blockIdx.y, sp = blockIdx.z;
  const int w = threadIdx.x >> 5, lane = threadIdx.x & 31;
  const int l16 = lane & 15, hi = lane >> 4;
  const int d = dc * 128 + w * 16 + l16;
  const float* arow = H + (size_t)(e * 16 + l16) * 4096;
  const float* wb = w2 + (size_t)e * 4096 * 1024;
  v8f acc = {};
  const int kbase = sp * 1024;
  for (int kk = 0; kk < 256; ++kk) {
    int k0 = kbase + kk * 4 + 2 * hi;
    v2f a = *(const v2f*)(arow + k0);
    v2f bv;
    bv.x = wb[(size_t)k0 * 1024 + d];
    bv.y = wb[(size_t)(k0 + 1) * 1024 + d];
    if (kk < 252) __builtin_prefetch(wb + (size_t)(k0 + 16) * 1024 + d, 0, 1);
    acc = wmma4(a, bv, acc);
  }
  float* out = part + (size_t)sp * (8 * 16 * 1024) + (size_t)e * 16 * 1024;
#pragma unroll
  for (int i = 0; i < 8; ++i)
    out[(size_t)(i + 8 * hi) * 1024 + d] = acc[i];
}

// Sum the 4 K-splits, add b2, relayout (e, r=2b+s) -> Y[b][j=2e+s][d].
__global__ void ybias_kernel(const float* __restrict__ part,
                             const float* __restrict__ b2,
                             float* __restrict__ Y) {
  int i = blockIdx.x * blockDim.x + threadIdx.x;  // 0..131071
  int e = i >> 14, rem = i & 16383, row = rem >> 10, d = rem & 1023;
  float v = part[i] + part[131072 + i] + part[262144 + i] + part[393216 + i] +
            b2[e * 1024 + d];
  int bb = row >> 1, j = e * 2 + (row & 1);
  Y[(size_t)(bb * 16 + j) * 1024 + d] = v;
}

// ---------------------------------------------------------------------------
// Kernel 8: out[b][n][d] = sum_j combine[b][n][j] * Y[b][j][d]  (K=16).
// ---------------------------------------------------------------------------
__global__ __launch_bounds__(256) void final_kernel(
    const float* __restrict__ combine, const float* __restrict__ Y,
    float* __restrict__ out) {
  const int b = blockIdx.x, tt = blockIdx.y, dc = blockIdx.z;
  const int w = threadIdx.x >> 5, lane = threadIdx.x & 31;
  const int l16 = lane & 15, hi = lane >> 4;
  const int dBase = dc * 128 + w * 16;
  const float* cb = combine + ((size_t)b * 4096 + tt * 16) * 16;
  const float* Yb = Y + (size_t)b * 16 * 1024;
  v8f acc = {};
#pragma unroll
  for (int kk = 0; kk < 4; ++kk) {
    int k0 = kk * 4 + 2 * hi;
    v2f a = *(const v2f*)(cb + l16 * 16 + k0);
    v2f bv;
    bv.x = Yb[(size_t)k0 * 1024 + dBase + l16];
    bv.y = Yb[(size_t)(k0 + 1) * 1024 + dBase + l16];
    acc = wmma4(a, bv, acc);
  }
  float* ob = out + ((size_t)b * 4096 + tt * 16) * 1024;
#pragma unroll
  for (int i = 0; i < 8; ++i)
    ob[(size_t)(i + 8 * hi) * 1024 + dBase + l16] = acc[i];
}

// ---------------------------------------------------------------------------
extern "C" void kernel_launch(void* const* d_in, const int* in_sizes, int n_in,
                              void* d_out, int out_size, void* d_ws,
                              size_t ws_size, hipStream_t stream) {
  (void)in_sizes; (void)n_in; (void)out_size; (void)ws_size;
  const float* x   = (const float*)d_in[0];
  // d_in[1] = attn_weight: unused by the reference computation
  const float* phi = (const float*)d_in[2];
  const float* kg  = (const float*)d_in[3];
  const float* kb  = (const float*)d_in[4];
  const float* qg  = (const float*)d_in[5];
  const float* qb  = (const float*)d_in[6];
  const float* lng = (const float*)d_in[7];
  const float* lnb = (const float*)d_in[8];
  const float* s0  = (const float*)d_in[9];
  const float* s1  = (const float*)d_in[10];
  const float* cw1 = (const float*)d_in[11];
  const float* cb1 = (const float*)d_in[12];
  const float* cw2 = (const float*)d_in[13];
  const float* cb2 = (const float*)d_in[14];
  // occ_* (d_in[15..18]) contribute exactly 0 in fp32 (see analysis): skipped.
  float* out = (float*)d_out;

  float* ws = (float*)d_ws;
  float* qB        = ws;                    // 16384
  float* dispatch  = qB + 16384;            // 524288
  float* combine   = dispatch + 524288;     // 524288
  float* slotsPart = combine + 524288;      // 4 * 131072
  float* slots     = slotsPart + 524288;    // 131072
  float* H         = slots + 131072;        // 524288
  float* Ypart     = H + 524288;            // 4 * 131072
  float* Y         = Ypart + 524288;        // 131072  (total ~11.6 MB)

  qnorm_kernel<<<16, 256, 0, stream>>>(phi, qg, qb, lng, lnb, qB);
  route_kernel<<<512, 128, 0, stream>>>(x, qB, kg, kb, s0, s1, dispatch,
                                        combine);
  slots_kernel<<<dim3(8, 8, 4), 256, 0, stream>>>(x, dispatch, slotsPart);
  sum4_kernel<<<512, 256, 0, stream>>>(slotsPart, slots, 131072);
  mlp1_kernel<<<dim3(8, 32), 256, 0, stream>>>(slots, cw1, cb1, H);
  mlp2_kernel<<<dim3(8, 8, 4), 256, 0, stream>>>(H, cw2, Ypart);
  ybias_kernel<<<512, 256, 0, stream>>>(Ypart, cb2, Y);
  final_kernel<<<dim3(8, 256, 8), 256, 0, stream>>>(combine, Y, out);
}